// RoPE_44641890074751
// MI455X (gfx1250) — compile-verified
//
#include <hip/hip_runtime.h>

// ---------------------------------------------------------------------------
// Multi-head "attention" (no softmax in reference => linear => associativity):
//   Q,K,V = x@W + b ; RoPE(Q), RoPE(K)
//   M_bh  = K_bh^T V_bh / 8            (64x64 per (b,h))
//   ctx   = Q @ M                      (block-diagonal per head)
//   out   = ctx @ Wo + bo
// All matrix math in fp32 via V_WMMA_F32_16X16X4_F32 (wave32).
// LDS tiles are stored in *fragment order*: for each k-group g (4 k-values),
// a lane's (k, k+1) pair is one contiguous float2 -> single ds_load_b64 per
// fragment, conflict-free banks (lane stride 16B; halves offset by 8B).
// ---------------------------------------------------------------------------

typedef __attribute__((ext_vector_type(2))) float v2f;
typedef __attribute__((ext_vector_type(8))) float v8f;

#define B_  2
#define L_  2048
#define D_  1024
#define H_  16
#define HD_ 64
#define BL_ (B_ * L_)

static __device__ __forceinline__ v8f wmma4(v2f a, v2f b, v8f c) {
    // 8 args: (neg_a, A, neg_b, B, c_mod, C, reuse_a, reuse_b)
    return __builtin_amdgcn_wmma_f32_16x16x4_f32(false, a, false, b,
                                                 (short)0, c, false, false);
}

// ---------------------------------------------------------------------------
// Generic fp32 GEMM + bias: C[M][N] = A[M][K] @ B[K][N] + bias[N]
// Block tile 128(M) x 128(N), BK = 16; 256 threads = 8 waves,
// each wave owns 32x64 (2 M-tiles x 4 N-tiles = 8 accumulators):
// per k-group: 2 A-frag + 4 B-frag b64 loads feed 8 WMMAs.
// Requires M%128==0, N%128==0, K%16==0 (true for all uses here).
// ---------------------------------------------------------------------------
__global__ __launch_bounds__(256) void gemm_bias_kernel(
    const float* __restrict__ A, const float* __restrict__ Bm,
    const float* __restrict__ bias, float* __restrict__ C,
    int M, int N, int K)
{
    __shared__ v2f lA[4][128][2];   // [kGroup][m][half] = {A[m][4g+2h], A[m][4g+2h+1]}
    __shared__ v2f lB[4][128][2];   // [kGroup][n][half] = {B[4g+2h][n], B[4g+2h+1][n]}
    float* fA = (float*)lA;
    float* fB = (float*)lB;

    const int tid  = threadIdx.x;
    const int wave = tid >> 5;
    const int lane = tid & 31;
    const int half = lane >> 4;
    const int lo   = lane & 15;
    const int mts  = (wave & 3) * 2;    // first of 2 M sub-tiles
    const int ntb  = (wave >> 2) * 4;   // first of 4 N sub-tiles

    const int mBase = blockIdx.y * 128;
    const int nBase = blockIdx.x * 128;

    v8f acc[2][4];
    #pragma unroll
    for (int mi = 0; mi < 2; ++mi)
        #pragma unroll
        for (int t = 0; t < 4; ++t)
            #pragma unroll
            for (int r = 0; r < 8; ++r) acc[mi][t][r] = 0.0f;

    for (int k0 = 0; k0 < K; k0 += 16) {
        #pragma unroll
        for (int j = 0; j < 8; ++j) {
            int i = tid + j * 256;
            {   // A tile: 128 x 16, consecutive tids read consecutive k
                int m = i >> 4, kk = i & 15;
                int g = kk >> 2, hh = (kk >> 1) & 1, p = kk & 1;
                fA[g * 512 + m * 4 + hh * 2 + p] =
                    A[(size_t)(mBase + m) * K + (k0 + kk)];
            }
            {   // B tile: 16 x 128, consecutive tids read consecutive n
                int kk = i >> 7, n = i & 127;
                int g = kk >> 2, hh = (kk >> 1) & 1, p = kk & 1;
                fB[g * 512 + n * 4 + hh * 2 + p] =
                    Bm[(size_t)(k0 + kk) * N + (nBase + n)];
            }
        }
        if (k0 + 16 < K) {  // hint next k-tile into cache (global_prefetch_b8)
            int m = tid >> 4, kk = tid & 15;
            __builtin_prefetch(&A[(size_t)(mBase + m) * K + (k0 + 16 + kk)], 0, 1);
            int kb = tid >> 7, n = tid & 127;
            __builtin_prefetch(&Bm[(size_t)(k0 + 16 + kb) * N + (nBase + n)], 0, 1);
        }
        __syncthreads();
        #pragma unroll
        for (int g = 0; g < 4; ++g) {
            v2f a0 = lA[g][(mts    ) * 16 + lo][half];
            v2f a1 = lA[g][(mts + 1) * 16 + lo][half];
            #pragma unroll
            for (int t = 0; t < 4; ++t) {
                v2f bf = lB[g][(ntb + t) * 16 + lo][half];
                acc[0][t] = wmma4(a0, bf, acc[0][t]);
                acc[1][t] = wmma4(a1, bf, acc[1][t]);
            }
        }
        __syncthreads();
    }

    #pragma unroll
    for (int mi = 0; mi < 2; ++mi) {
        #pragma unroll
        for (int t = 0; t < 4; ++t) {
            int col = nBase + (ntb + t) * 16 + lo;
            float bv = bias[col];
            #pragma unroll
            for (int r = 0; r < 8; ++r) {
                int row = mBase + (mts + mi) * 16 + r + 8 * half;
                C[(size_t)row * N + col] = acc[mi][t][r] + bv;
            }
        }
    }
}

// ---------------------------------------------------------------------------
// RoPE (reference uses freqs = 10000^{-scale}, scale = 0,2,...,62).
// Pair (h*64 + 2j, h*64 + 2j + 1) == flat pair index, so pairs are contiguous.
// Applies in place to both Q and K.
// ---------------------------------------------------------------------------
__global__ void rope_kernel(float* __restrict__ q, float* __restrict__ k)
{
    int idx = blockIdx.x * blockDim.x + threadIdx.x;     // pair index
    if (idx >= BL_ * (D_ / 2)) return;
    int row = idx >> 9;                                  // / (D_/2)
    int pr  = idx & 511;
    int j   = pr & 31;                                   // pair within head
    float pos  = (float)(row & (L_ - 1));                // l position
    float freq = powf(10000.0f, -2.0f * (float)j);
    float ang  = pos * freq;
    float c = cosf(ang), s = sinf(ang);

    float2* qp = (float2*)q;
    float2* kp = (float2*)k;
    float2 v = qp[idx];
    qp[idx] = make_float2(v.x * c - v.y * s, v.x * s + v.y * c);
    v = kp[idx];
    kp[idx] = make_float2(v.x * c - v.y * s, v.x * s + v.y * c);
}

// ---------------------------------------------------------------------------
// M_bh = (1/8) * K_bh^T @ V_bh : [64,2048]^T x [2048,64] -> [64,64]
// One block per (b,h); 8 waves x 2 output tiles; K-dim (= l) staged via LDS
// in fragment order: lK[g][m][half] = {K[l=4g+2h][m], K[l=4g+2h+1][m]}.
// ---------------------------------------------------------------------------
__global__ __launch_bounds__(256) void ktv_kernel(
    const float* __restrict__ Kq, const float* __restrict__ V,
    float* __restrict__ M64)
{
    const int b = blockIdx.x >> 4;
    const int h = blockIdx.x & 15;
    const float* kbase = Kq + (size_t)b * L_ * D_ + h * HD_;
    const float* vbase = V  + (size_t)b * L_ * D_ + h * HD_;

    __shared__ v2f lK[4][HD_][2];
    __shared__ v2f lV[4][HD_][2];
    float* fK = (float*)lK;
    float* fV = (float*)lV;

    const int tid  = threadIdx.x;
    const int wave = tid >> 5;
    const int lane = tid & 31;
    const int half = lane >> 4;
    const int lo   = lane & 15;
    const int mt   = wave & 3;
    const int ntb  = (wave >> 2) * 2;

    v8f acc[2];
    #pragma unroll
    for (int t = 0; t < 2; ++t)
        #pragma unroll
        for (int r = 0; r < 8; ++r) acc[t][r] = 0.0f;

    for (int l0 = 0; l0 < L_; l0 += 16) {
        #pragma unroll
        for (int j = 0; j < 4; ++j) {
            int i = tid + j * 256;
            int ll = i >> 6, m = i & 63;
            int g = ll >> 2, hh = (ll >> 1) & 1, p = ll & 1;
            int off = g * 256 + m * 4 + hh * 2 + p;
            fK[off] = kbase[(size_t)(l0 + ll) * D_ + m];
            fV[off] = vbase[(size_t)(l0 + ll) * D_ + m];
        }
        __syncthreads();
        #pragma unroll
        for (int g = 0; g < 4; ++g) {
            v2f a = lK[g][mt * 16 + lo][half];
            #pragma unroll
            for (int t = 0; t < 2; ++t) {
                v2f bf = lV[g][(ntb + t) * 16 + lo][half];
                acc[t] = wmma4(a, bf, acc[t]);
            }
        }
        __syncthreads();
    }

    float* out = M64 + (size_t)blockIdx.x * HD_ * HD_;
    #pragma unroll
    for (int t = 0; t < 2; ++t) {
        int n = (ntb + t) * 16 + lo;
        #pragma unroll
        for (int r = 0; r < 8; ++r) {
            int m = mt * 16 + r + 8 * half;
            out[m * HD_ + n] = acc[t][r] * 0.125f;   // 1/sqrt(64)
        }
    }
}

// ---------------------------------------------------------------------------
// ctx[b, l, h*64+n] = Q[b, l, h*64+:] @ M_bh   — [2048,64] x [64,64] per (b,h)
// grid = (L/64, H, B); 64x64 output tile per block, whole K (=64) in LDS,
// fragment-order layout as above.
// ---------------------------------------------------------------------------
__global__ __launch_bounds__(256) void qm_kernel(
    const float* __restrict__ Q, const float* __restrict__ M64,
    float* __restrict__ Ctx)
{
    const int lt = blockIdx.x, h = blockIdx.y, b = blockIdx.z;
    const float* qbase = Q   + (size_t)b * L_ * D_ + h * HD_;
    const float* mbase = M64 + (size_t)(b * H_ + h) * HD_ * HD_;
    float*       cbase = Ctx + (size_t)b * L_ * D_ + h * HD_;

    __shared__ v2f lQ[16][HD_][2];   // [kGroup][l][half]
    __shared__ v2f lM[16][HD_][2];   // [kGroup][n][half]
    float* fQ = (float*)lQ;
    float* fM = (float*)lM;

    const int tid  = threadIdx.x;
    const int wave = tid >> 5;
    const int lane = tid & 31;
    const int half = lane >> 4;
    const int lo   = lane & 15;
    const int mt   = wave & 3;
    const int ntb  = (wave >> 2) * 2;

    #pragma unroll
    for (int j = 0; j < 16; ++j) {
        int i = tid + j * 256;
        {   // Q tile: [l=64][k=64]
            int ll = i >> 6, kc = i & 63;
            int g = kc >> 2, hh = (kc >> 1) & 1, p = kc & 1;
            fQ[g * 256 + ll * 4 + hh * 2 + p] =
                qbase[(size_t)(lt * 64 + ll) * D_ + kc];
        }
        {   // M tile: [k=64][n=64], contiguous in memory
            int kc = i >> 6, n = i & 63;
            int g = kc >> 2, hh = (kc >> 1) & 1, p = kc & 1;
            fM[g * 256 + n * 4 + hh * 2 + p] = mbase[i];
        }
    }
    __syncthreads();

    v8f acc[2];
    #pragma unroll
    for (int t = 0; t < 2; ++t)
        #pragma unroll
        for (int r = 0; r < 8; ++r) acc[t][r] = 0.0f;

    #pragma unroll
    for (int g = 0; g < 16; ++g) {
        v2f a = lQ[g][mt * 16 + lo][half];
        #pragma unroll
        for (int t = 0; t < 2; ++t) {
            v2f bf = lM[g][(ntb + t) * 16 + lo][half];
            acc[t] = wmma4(a, bf, acc[t]);
        }
    }

    #pragma unroll
    for (int t = 0; t < 2; ++t) {
        int n = (ntb + t) * 16 + lo;
        #pragma unroll
        for (int r = 0; r < 8; ++r) {
            int row = lt * 64 + mt * 16 + r + 8 * half;
            cbase[(size_t)row * D_ + n] = acc[t][r];
        }
    }
}

// ---------------------------------------------------------------------------
extern "C" void kernel_launch(void* const* d_in, const int* in_sizes, int n_in,
                              void* d_out, int out_size, void* d_ws, size_t ws_size,
                              hipStream_t stream)
{
    (void)in_sizes; (void)n_in; (void)out_size; (void)ws_size;
    const float* x  = (const float*)d_in[0];
    const float* Wq = (const float*)d_in[1];
    const float* bq = (const float*)d_in[2];
    const float* Wk = (const float*)d_in[3];
    const float* bk = (const float*)d_in[4];
    const float* Wv = (const float*)d_in[5];
    const float* bv = (const float*)d_in[6];
    const float* Wo = (const float*)d_in[7];
    const float* bo = (const float*)d_in[8];
    float* out = (float*)d_out;

    const size_t mat = (size_t)BL_ * D_;        // 4096*1024 floats = 16 MB
    float* Q   = (float*)d_ws;
    float* Kr  = Q  + mat;
    float* V   = Kr + mat;
    float* M64 = V  + mat;                      // 32 * 64 * 64 floats
    float* Ctx = Kr;                            // K is dead after ktv -> reuse

    dim3 gemmGrid(D_ / 128, BL_ / 128);         // (8, 32)

    gemm_bias_kernel<<<gemmGrid, 256, 0, stream>>>(x, Wq, bq, Q,  BL_, D_, D_);
    gemm_bias_kernel<<<gemmGrid, 256, 0, stream>>>(x, Wk, bk, Kr, BL_, D_, D_);
    gemm_bias_kernel<<<gemmGrid, 256, 0, stream>>>(x, Wv, bv, V,  BL_, D_, D_);

    int pairs = BL_ * (D_ / 2);
    rope_kernel<<<(pairs + 255) / 256, 256, 0, stream>>>(Q, Kr);

    ktv_kernel<<<B_ * H_, 256, 0, stream>>>(Kr, V, M64);

    qm_kernel<<<dim3(L_ / 64, H_, B_), 256, 0, stream>>>(Q, M64, Ctx);

    gemm_bias_kernel<<<gemmGrid, 256, 0, stream>>>(Ctx, Wo, bo, out, BL_, D_, D_);
}